// Codebook_56100862820665
// MI455X (gfx1250) — compile-verified
//
#include <hip/hip_runtime.h>

typedef __attribute__((ext_vector_type(2))) float v2f;
typedef __attribute__((ext_vector_type(8))) float v8f;

#define DIM 64
#define TILE_K 128          // codebook rows staged in LDS per iteration
#define ROWS_PER_WG 128     // 8 waves * 16 rows each
#define LDS_STRIDE 68       // 64 + 4 pad: conflict-free ds_load_b64 B-frag reads

// ---------- pass 1: squared norms of codebook rows ----------
__global__ void vq_c2_kernel(const float* __restrict__ cb,
                             float* __restrict__ c2, int K) {
    int k = blockIdx.x * blockDim.x + threadIdx.x;
    if (k < K) {
        const float4* row = (const float4*)(cb + (size_t)k * DIM);
        float s = 0.f;
#pragma unroll
        for (int i = 0; i < DIM / 4; ++i) {
            float4 v = row[i];
            s += v.x * v.x + v.y * v.y + v.z * v.z + v.w * v.w;
        }
        c2[k] = s;
    }
}

// ---------- pass 2: WMMA distance + argmin + gather ----------
__global__ __launch_bounds__(256)
void vq_argmin_kernel(const float* __restrict__ z,
                      const float* __restrict__ cb,
                      const float* __restrict__ c2,
                      float* __restrict__ zq,
                      int* __restrict__ idx_out,
                      int K) {
    __shared__ float cb_lds[TILE_K * LDS_STRIDE];
    __shared__ float c2_lds[TILE_K];
    __shared__ int   idx_lds[ROWS_PER_WG];

    const int tid   = threadIdx.x;
    const int lane  = tid & 31;
    const int wave  = tid >> 5;
    const int m     = lane & 15;   // row-within-strip (A) / code-column (B)
    const int hi    = lane >> 4;   // K-pair select: 0 -> {0,1}, 1 -> {2,3}
    const int rowBase = blockIdx.x * ROWS_PER_WG;
    const int myRow   = rowBase + wave * 16 + m;

    // A fragments: this wave's 16 z rows, 16x4 f32 layout, kept in registers.
    v2f afrag[16];
    {
        const float* zr = z + (size_t)myRow * DIM + hi * 2;
#pragma unroll
        for (int t = 0; t < 16; ++t)
            afrag[t] = *(const v2f*)(zr + t * 4);
    }

    float bestS[8];
    int   bestI[8];
#pragma unroll
    for (int v = 0; v < 8; ++v) { bestS[v] = 3.4e38f; bestI[v] = 0; }

    for (int k0 = 0; k0 < K; k0 += TILE_K) {
        // Stage codebook tile + c2 tile into LDS (256 threads x 8 float4).
#pragma unroll
        for (int i = 0; i < 8; ++i) {
            int item = tid + i * 256;          // 0..2047
            int r    = item >> 4;              // code row in tile
            int c4   = item & 15;              // float4 column
            float4 val = *(const float4*)(cb + (size_t)(k0 + r) * DIM + c4 * 4);
            *(float4*)(&cb_lds[r * LDS_STRIDE + c4 * 4]) = val;
        }
        if (tid < TILE_K) c2_lds[tid] = c2[k0 + tid];
        __syncthreads();

        // Prefetch next tile toward L2 while we compute (global_prefetch_b8).
        if (k0 + TILE_K < K)
            __builtin_prefetch(cb + (size_t)(k0 + TILE_K) * DIM + tid * 8, 0, 1);

#pragma unroll
        for (int sub = 0; sub < TILE_K / 16; ++sub) {
            const float* brow = &cb_lds[(sub * 16 + m) * LDS_STRIDE + hi * 2];
            v8f acc = {};
#pragma unroll
            for (int t = 0; t < 16; ++t) {
                v2f b = *(const v2f*)(brow + t * 4);
                acc = __builtin_amdgcn_wmma_f32_16x16x4_f32(
                    /*neg_a=*/false, afrag[t], /*neg_b=*/false, b,
                    /*c_mod=*/(short)0, acc, /*reuse_a=*/false, /*reuse_b=*/false);
            }
            const int   kg = k0 + sub * 16 + m;   // code column this lane owns
            const float cc = c2_lds[sub * 16 + m];
#pragma unroll
            for (int v = 0; v < 8; ++v) {
                float s = cc - 2.0f * acc[v];     // ||c||^2 - 2 x.c  (x^2 dropped)
                if (s < bestS[v]) { bestS[v] = s; bestI[v] = kg; }
            }
        }
        __syncthreads();
    }

    // Min-reduce across the 16 lanes holding different code columns.
#pragma unroll
    for (int off = 8; off >= 1; off >>= 1) {
#pragma unroll
        for (int v = 0; v < 8; ++v) {
            float os = __shfl_xor(bestS[v], off, 32);
            int   oi = __shfl_xor(bestI[v], off, 32);
            if (os < bestS[v] || (os == bestS[v] && oi < bestI[v])) {
                bestS[v] = os; bestI[v] = oi;
            }
        }
    }

    // Lane m==0 of each half-wave owns rows (wave*16 + hi*8 + v).
    if (m == 0) {
#pragma unroll
        for (int v = 0; v < 8; ++v) {
            int r = wave * 16 + hi * 8 + v;
            idx_lds[r] = bestI[v];
            idx_out[rowBase + r] = bestI[v];
        }
    }
    __syncthreads();

    // Gather z_q rows from the (L2-hot) codebook; 2 threads per row.
    {
        int r = tid >> 1;
        int h = tid & 1;
        int code = idx_lds[r];
        const float4* src = (const float4*)(cb + (size_t)code * DIM) + h * 8;
        float4* dst = (float4*)(zq + (size_t)(rowBase + r) * DIM) + h * 8;
#pragma unroll
        for (int i = 0; i < 8; ++i) dst[i] = src[i];
    }
}

extern "C" void kernel_launch(void* const* d_in, const int* in_sizes, int n_in,
                              void* d_out, int out_size, void* d_ws, size_t ws_size,
                              hipStream_t stream) {
    const float* z  = (const float*)d_in[0];   // [B,T,D] f32
    const float* cb = (const float*)d_in[1];   // [K,D]   f32
    const int N = in_sizes[0] / DIM;           // B*T
    const int K = in_sizes[1] / DIM;

    float* c2      = (float*)d_ws;             // K floats of scratch
    float* zq      = (float*)d_out;            // N*D floats
    int*   idx_out = (int*)d_out + (size_t)N * DIM;  // N int32 after z_q

    vq_c2_kernel<<<(K + 255) / 256, 256, 0, stream>>>(cb, c2, K);
    vq_argmin_kernel<<<N / ROWS_PER_WG, 256, 0, stream>>>(z, cb, c2, zq, idx_out, K);
}